// GraphNet_64725157151179
// MI455X (gfx1250) — compile-verified
//
#include <hip/hip_runtime.h>

typedef __attribute__((ext_vector_type(2))) float v2f;
typedef __attribute__((ext_vector_type(8))) float v8f;

#define BN_EPS 1e-5f

// ---------------------------------------------------------------------------
// h[i] = elem_emb[x[i,0]] + chir_emb[x[i,1]]        (one thread = 4 channels)
// ---------------------------------------------------------------------------
__global__ __launch_bounds__(256)
void k_node_embed(const int* __restrict__ x,
                  const float* __restrict__ elem_emb,
                  const float* __restrict__ chir_emb,
                  float* __restrict__ h, int n)
{
    int t = blockIdx.x * blockDim.x + threadIdx.x;
    if (t >= n * 32) return;
    int node = t >> 5;
    int c4   = t & 31;
    int e = x[2 * node];
    int c = x[2 * node + 1];
    float4 ev = ((const float4*)elem_emb)[(size_t)e * 32 + c4];
    float4 cv = ((const float4*)chir_emb)[(size_t)c * 32 + c4];
    float4 r = make_float4(ev.x + cv.x, ev.y + cv.y, ev.z + cv.z, ev.w + cv.w);
    ((float4*)h)[(size_t)node * 32 + c4] = r;
}

// ---------------------------------------------------------------------------
// agg[dst] += relu(h[src] + bond_type_emb[a0] + bond_dir_emb[a1])
// edge embeddings recomputed on the fly (tables are tiny, L2-resident).
// One thread = one edge x 4 channels; 32 consecutive threads cover one edge
// (512B coalesced float4 accesses). agg must be pre-initialized to h (=> z).
// ---------------------------------------------------------------------------
__global__ __launch_bounds__(256)
void k_scatter(const int* __restrict__ edge_index,
               const int* __restrict__ edge_attr,
               const float* __restrict__ bt_emb,
               const float* __restrict__ bd_emb,
               const float* __restrict__ h,
               float* __restrict__ agg, int E)
{
    int t = blockIdx.x * blockDim.x + threadIdx.x;
    if (t >= E * 32) return;
    int e  = t >> 5;
    int c4 = t & 31;
    int src = edge_index[e];
    int dst = edge_index[E + e];
    int bt  = edge_attr[2 * e];
    int bd  = edge_attr[2 * e + 1];
    float4 hv = ((const float4*)h)[(size_t)src * 32 + c4];
    float4 bv = ((const float4*)bt_emb)[(size_t)bt * 32 + c4];
    float4 dv = ((const float4*)bd_emb)[(size_t)bd * 32 + c4];
    float4 m;
    m.x = fmaxf(hv.x + bv.x + dv.x, 0.f);
    m.y = fmaxf(hv.y + bv.y + dv.y, 0.f);
    m.z = fmaxf(hv.z + bv.z + dv.z, 0.f);
    m.w = fmaxf(hv.w + bv.w + dv.w, 0.f);
    float* base = agg + (size_t)dst * 128 + c4 * 4;
    atomicAdd(base + 0, m.x);
    atomicAdd(base + 1, m.y);
    atomicAdd(base + 2, m.z);
    atomicAdd(base + 3, m.w);
}

__global__ void k_zero_stats(float* __restrict__ stats)
{
    stats[threadIdx.x] = 0.f;   // 256 threads, 256 floats
}

// ---------------------------------------------------------------------------
// Fused MLP: out = relu(z @ W1 + b1) @ W2 + b2 for a 16-row block, hidden
// tile (16x256) kept entirely in LDS. fp32 WMMA 16x16x4, wave32, 8 waves/WG.
// Also accumulates per-channel sum / sum^2 for BatchNorm into stats[256].
// LDS rows padded (132 / 260 floats) so column reads across 16 rows hit
// 16 distinct banks (stride 4 mod 64).
// ---------------------------------------------------------------------------
__global__ __launch_bounds__(256)
void k_mlp(const float* __restrict__ z,
           const float* __restrict__ W1, const float* __restrict__ b1,
           const float* __restrict__ W2, const float* __restrict__ b2,
           float* __restrict__ out, float* __restrict__ stats, int n)
{
    __shared__ float zb[16][132];   // z tile, later reused as out tile
    __shared__ float hb[16][260];   // hidden tile (16 x 256)

    const int tid  = threadIdx.x;
    const int wave = tid >> 5;          // 0..7
    const int lane = tid & 31;
    const int m0   = blockIdx.x << 4;   // first row of this block
    const int row  = lane & 15;         // fragment row / col index
    const int kh   = (lane >> 4) << 1;  // K offset: 0 (lanes 0-15) or 2 (16-31)
    const int rb   = (lane >> 4) << 3;  // C/D row base: 0 or 8

    // --- stage z tile (16 x 128) into LDS ---
    for (int i = tid; i < 16 * 32; i += 256) {
        int r = i >> 5, c4 = i & 31;
        float4 v = (m0 + r < n) ? ((const float4*)z)[(size_t)(m0 + r) * 32 + c4]
                                : make_float4(0.f, 0.f, 0.f, 0.f);
        zb[r][c4 * 4 + 0] = v.x;
        zb[r][c4 * 4 + 1] = v.y;
        zb[r][c4 * 4 + 2] = v.z;
        zb[r][c4 * 4 + 3] = v.w;
    }
    __syncthreads();

    // --- GEMM1: hidden = relu(z @ W1 + b1); each wave owns 2 col-tiles ---
    for (int t = 0; t < 2; ++t) {
        const int nb = ((wave << 1) + t) << 4;      // hidden col base (0..240)
        v8f acc = {};
        for (int k0 = 0; k0 < 128; k0 += 4) {
            v2f a, b;
            a.x = zb[row][k0 + kh];
            a.y = zb[row][k0 + kh + 1];
            b.x = W1[(size_t)(k0 + kh) * 256 + nb + row];
            b.y = W1[(size_t)(k0 + kh + 1) * 256 + nb + row];
            acc = __builtin_amdgcn_wmma_f32_16x16x4_f32(false, a, false, b,
                                                        (short)0, acc, false, false);
        }
        const int col  = nb + row;
        const float bs = b1[col];
        #pragma unroll
        for (int v = 0; v < 8; ++v) {
            float val = acc[v] + bs;
            hb[rb + v][col] = val > 0.f ? val : 0.f;
        }
    }
    __syncthreads();

    // --- GEMM2: out = hidden @ W2 + b2; each wave owns 1 col-tile ---
    {
        const int nb = wave << 4;                   // out col base (0..112)
        v8f acc = {};
        for (int k0 = 0; k0 < 256; k0 += 4) {
            v2f a, b;
            a.x = hb[row][k0 + kh];
            a.y = hb[row][k0 + kh + 1];
            b.x = W2[(size_t)(k0 + kh) * 128 + nb + row];
            b.y = W2[(size_t)(k0 + kh + 1) * 128 + nb + row];
            acc = __builtin_amdgcn_wmma_f32_16x16x4_f32(false, a, false, b,
                                                        (short)0, acc, false, false);
        }
        const int col  = nb + row;
        const float bs = b2[col];
        #pragma unroll
        for (int v = 0; v < 8; ++v) {
            int r = rb + v;
            float val = acc[v] + bs;
            zb[r][col] = val;                       // reuse zb as out tile
            if (m0 + r < n) out[(size_t)(m0 + r) * 128 + col] = val;
        }
    }
    __syncthreads();

    // --- BatchNorm partial reduction: thread c handles channel c ---
    if (tid < 128) {
        int rmax = n - m0; if (rmax > 16) rmax = 16;
        float s = 0.f, s2 = 0.f;
        for (int r = 0; r < rmax; ++r) {
            float v = zb[r][tid];
            s += v; s2 += v * v;
        }
        atomicAdd(&stats[tid],       s);
        atomicAdd(&stats[128 + tid], s2);
    }
}

// ---------------------------------------------------------------------------
// stats -> per-channel scale/shift   (1 block, 128 threads)
// ---------------------------------------------------------------------------
__global__ void k_bn_final(const float* __restrict__ stats,
                           const float* __restrict__ gamma,
                           const float* __restrict__ beta,
                           float* __restrict__ ss, float inv_n)
{
    int c = threadIdx.x;
    float mean = stats[c] * inv_n;
    float var  = stats[128 + c] * inv_n - mean * mean;   // biased variance
    float sc   = gamma[c] * rsqrtf(var + BN_EPS);
    ss[c]       = sc;
    ss[128 + c] = beta[c] - mean * sc;
}

// ---------------------------------------------------------------------------
// dst = (relu?) (out * scale + shift)        (one thread = 4 channels)
// ---------------------------------------------------------------------------
__global__ __launch_bounds__(256)
void k_bn_apply(const float* __restrict__ out,
                const float* __restrict__ ss,
                float* __restrict__ dst, int n, int do_relu)
{
    int t = blockIdx.x * blockDim.x + threadIdx.x;
    if (t >= n * 32) return;
    int c4 = (t & 31) << 2;
    float4 v = ((const float4*)out)[t];
    float4 r;
    r.x = v.x * ss[c4 + 0] + ss[128 + c4 + 0];
    r.y = v.y * ss[c4 + 1] + ss[128 + c4 + 1];
    r.z = v.z * ss[c4 + 2] + ss[128 + c4 + 2];
    r.w = v.w * ss[c4 + 3] + ss[128 + c4 + 3];
    if (do_relu) {
        r.x = fmaxf(r.x, 0.f); r.y = fmaxf(r.y, 0.f);
        r.z = fmaxf(r.z, 0.f); r.w = fmaxf(r.w, 0.f);
    }
    ((float4*)dst)[t] = r;
}

// ---------------------------------------------------------------------------
extern "C" void kernel_launch(void* const* d_in, const int* in_sizes, int n_in,
                              void* d_out, int out_size, void* d_ws, size_t ws_size,
                              hipStream_t stream)
{
    const int*   x          = (const int*)d_in[0];
    const int*   edge_index = (const int*)d_in[1];
    const int*   edge_attr  = (const int*)d_in[2];
    const float* elem_emb   = (const float*)d_in[3];
    const float* chir_emb   = (const float*)d_in[4];
    const float* bt_emb     = (const float*)d_in[5];
    const float* bd_emb     = (const float*)d_in[6];
    const float* W1         = (const float*)d_in[7];
    const float* b1         = (const float*)d_in[8];
    const float* W2         = (const float*)d_in[9];
    const float* b2         = (const float*)d_in[10];
    const float* bn_gamma   = (const float*)d_in[11];
    const float* bn_beta    = (const float*)d_in[12];

    const int n          = in_sizes[0] / 2;     // 50000 nodes
    const int E          = in_sizes[1] / 2;     // 600000 edges
    const int num_layers = in_sizes[11] / 128;  // 5

    // workspace layout (floats): h | z/agg | mlp_out | stats[256] | ss[256]
    float* ws    = (float*)d_ws;
    float* h     = ws;
    float* agg   = h     + (size_t)n * 128;
    float* mlp   = agg   + (size_t)n * 128;
    float* stats = mlp   + (size_t)n * 128;
    float* ss    = stats + 256;

    const dim3 blk(256);
    const int gN = (n * 32 + 255) / 256;        // node-channel grid
    const int gE = (E * 32 + 255) / 256;        // edge-channel grid
    const int gM = (n + 15) / 16;               // 16-row MLP blocks

    k_node_embed<<<gN, blk, 0, stream>>>(x, elem_emb, chir_emb, h, n);

    for (int layer = 0; layer < num_layers; ++layer) {
        // z := h, then scatter-add messages into it
        hipMemcpyAsync(agg, h, (size_t)n * 128 * sizeof(float),
                       hipMemcpyDeviceToDevice, stream);
        k_scatter<<<gE, blk, 0, stream>>>(edge_index, edge_attr,
                                          bt_emb, bd_emb, h, agg, E);
        k_zero_stats<<<1, 256, 0, stream>>>(stats);
        k_mlp<<<gM, blk, 0, stream>>>(agg, W1, b1, W2, b2, mlp, stats, n);
        k_bn_final<<<1, 128, 0, stream>>>(stats, bn_gamma + layer * 128,
                                          bn_beta + layer * 128, ss, 1.0f / n);
        float* dst = (layer == num_layers - 1) ? (float*)d_out : h;
        k_bn_apply<<<gN, blk, 0, stream>>>(mlp, ss, dst, n,
                                           layer < num_layers - 1 ? 1 : 0);
    }
}